// LinearEquations_54073638256982
// MI455X (gfx1250) — compile-verified
//
#include <hip/hip_runtime.h>

// ---------------------------------------------------------------------------
// x1[o,n,c] = sum_p X[n,p] * (dags*W)[o,p,c]   (O=8, N=200000, P=C=64)
// reg[o]    = 1e-3 * sum(W[o]^2)
// out       = concat(x1 (O*N*64), dags (O*64*64), reg (O))
// Store-bound (~460 MB @ 23.3 TB/s => ~20us floor); f32 math preserved via
// V_WMMA_F32_16X16X4_F32. X read from HBM exactly once (A frags held in
// registers, reused across all 8 o and 4 c-tiles).
// ---------------------------------------------------------------------------

typedef __attribute__((ext_vector_type(2))) float v2f;
typedef __attribute__((ext_vector_type(4))) float v4f;
typedef __attribute__((ext_vector_type(8))) float v8f;

#define NROWS 200000
#define DDIM  64
#define ODIM  8
#define OPC   (ODIM * DDIM * DDIM)   // 32768 elements per dags/W
#define BLOCK_ROWS 128               // 8 waves * 16 rows
#define XGS 34    // sX group stride (floats) per (row,h): conflict-free v2f frags
#define WGS 160   // sW group stride (floats) per p-pair:  32h bank split for b64

// ---- prep 1: WM = dags * W  (into workspace), and copy dags into output ----
__global__ __launch_bounds__(256) void prep_wm_kernel(
    const float* __restrict__ dags, const float* __restrict__ W,
    float* __restrict__ wm, float* __restrict__ out_dags) {
  int i = blockIdx.x * 256 + threadIdx.x;
  if (i < OPC) {
    float d = dags[i];
    wm[i] = d * W[i];
    out_dags[i] = d;
  }
}

// ---- prep 2: reg[o] = 1e-3 * sum(W[o]^2) ----
__global__ __launch_bounds__(256) void prep_reg_kernel(
    const float* __restrict__ W, float* __restrict__ out_reg) {
  __shared__ float red[256];
  const int o = blockIdx.x;
  const float* w = W + o * (DDIM * DDIM);
  float s = 0.f;
  for (int j = threadIdx.x; j < DDIM * DDIM; j += 256) {
    float v = w[j];
    s += v * v;
  }
  red[threadIdx.x] = s;
  __syncthreads();
  for (int off = 128; off > 0; off >>= 1) {
    if (threadIdx.x < off) red[threadIdx.x] += red[threadIdx.x + off];
    __syncthreads();
  }
  if (threadIdx.x == 0) out_reg[o] = 0.001f * red[0];
}

// ---- main batched GEMM: block = 128 N-rows, loops o; 4 c-tiles per wave ----
__global__ __launch_bounds__(256) void gemm_wmma_kernel(
    const float* __restrict__ X, const float* __restrict__ WM,
    float* __restrict__ x1) {
  // sX[row][h][2*ts + j] = X[row][4*ts + 2*h + j]  -> A frag = contiguous v2f
  __shared__ float sX[BLOCK_ROWS * 2 * XGS];  // 34,816 B
  // sW[g][2*c + j] = WM[2*g + j][c], g = 2*ts + h -> B frag = contiguous v2f
  __shared__ float sW[32 * WGS];              // 20,480 B

  const int t    = threadIdx.x;
  const int lane = t & 31;
  const int wid  = t >> 5;     // wave id 0..7
  const int lm   = lane & 15;  // M (A/D rows) or N (B/D cols) index
  const int h    = lane >> 4;  // lane-half: selects K pair / row-half
  const int nblk = blockIdx.x * BLOCK_ROWS;

  // Stage X tile (128 x 64 f32): coalesced b128 loads -> swizzled LDS.
  #pragma unroll
  for (int i = 0; i < 8; ++i) {
    int f   = i * 256 + t;        // float4 index within tile
    int row = f >> 4;
    int ts  = f & 15;             // K-step this float4 covers (cols 4ts..4ts+3)
    if (nblk + row < NROWS) {
      v4f v = *(const v4f*)(X + (size_t)(nblk + row) * DDIM + 4 * ts);
      v2f lo = {v[0], v[1]};      // K = 4ts, 4ts+1  -> h = 0
      v2f hi = {v[2], v[3]};      // K = 4ts+2,+3    -> h = 1
      *(v2f*)&sX[(row * 2 + 0) * XGS + 2 * ts] = lo;
      *(v2f*)&sX[(row * 2 + 1) * XGS + 2 * ts] = hi;
    }
  }
  __syncthreads();

  // Preload this wave's 16 A fragments once (16 rows x K=64).
  // A layout (16x4 f32): lanes 0-15 = M=0..15; VGPR0/1 hold K = 2h, 2h+1.
  const int  n0     = nblk + wid * 16;
  const bool active = (n0 < NROWS);  // wave-uniform; N % 16 == 0
  v2f afrag[16];
  if (active) {
    const float* xrow = &sX[((wid * 16 + lm) * 2 + h) * XGS];
    #pragma unroll
    for (int ts = 0; ts < 16; ++ts)
      afrag[ts] = *(const v2f*)&xrow[2 * ts];
  }

  for (int o = 0; o < ODIM; ++o) {
    __syncthreads();
    // Stage WM[o] (64 x 64 f32) pair-interleaved: sW[g][2c+j] = WM[2g+j][c].
    {
      const float* src = WM + o * (DDIM * DDIM);
      #pragma unroll
      for (int i = 0; i < 4; ++i) {
        int f  = i * 256 + t;     // 0..1023
        int g  = f >> 5;          // p-pair index 0..31
        int cp = f & 31;          // column pair 0..31 (covers c = 2cp, 2cp+1)
        v2f r0 = *(const v2f*)(src + (2 * g + 0) * DDIM + 2 * cp);
        v2f r1 = *(const v2f*)(src + (2 * g + 1) * DDIM + 2 * cp);
        v4f q  = {r0[0], r1[0], r0[1], r1[1]};
        *(v4f*)&sW[g * WGS + 4 * cp] = q;   // 16B aligned
      }
    }
    __syncthreads();
    if (o + 1 < ODIM)  // overlap next WM fetch with compute (global_prefetch)
      __builtin_prefetch(WM + (o + 1) * (DDIM * DDIM) + t * 16, 0, 0);

    if (active) {
      #pragma unroll
      for (int ct = 0; ct < 4; ++ct) {
        const int c0 = ct * 16;
        v8f acc = {0.f, 0.f, 0.f, 0.f, 0.f, 0.f, 0.f, 0.f};
        #pragma unroll
        for (int ts = 0; ts < 16; ++ts) {
          const int g = 2 * ts + h;
          // Single aligned ds_load_b64: both K values for this lane, no moves.
          v2f b = *(const v2f*)&sW[g * WGS + 2 * (c0 + lm)];
          acc = __builtin_amdgcn_wmma_f32_16x16x4_f32(
              /*neg_a=*/false, afrag[ts], /*neg_b=*/false, b,
              /*c_mod=*/(short)0, acc, /*reuse_a=*/false, /*reuse_b=*/false);
        }
        // D layout: lanes 0-15 VGPR r -> M=r; lanes 16-31 -> M=r+8.
        // Each half-wave store is a contiguous 64B segment.
        float* dst = x1 + ((size_t)o * NROWS + (size_t)(n0 + 8 * h)) * DDIM
                        + c0 + lm;
        #pragma unroll
        for (int r = 0; r < 8; ++r)
          dst[(size_t)r * DDIM] = acc[r];
      }
    }
  }
}

extern "C" void kernel_launch(void* const* d_in, const int* in_sizes, int n_in,
                              void* d_out, int out_size, void* d_ws,
                              size_t ws_size, hipStream_t stream) {
  const float* X    = (const float*)d_in[0];
  const float* dags = (const float*)d_in[1];
  const float* W    = (const float*)d_in[2];

  float* out      = (float*)d_out;
  float* x1       = out;                                  // O*N*64 floats
  float* out_dags = out + (size_t)ODIM * NROWS * DDIM;    // +102,400,000
  float* out_reg  = out_dags + OPC;                       // +32,768
  float* wm       = (float*)d_ws;                         // 128 KB scratch

  prep_wm_kernel<<<(OPC + 255) / 256, 256, 0, stream>>>(dags, W, wm, out_dags);
  prep_reg_kernel<<<ODIM, 256, 0, stream>>>(W, out_reg);
  gemm_wmma_kernel<<<(NROWS + BLOCK_ROWS - 1) / BLOCK_ROWS, 256, 0, stream>>>(
      X, wm, x1);
}